// CentersDistance_74801150427724
// MI455X (gfx1250) — compile-verified
//
#include <hip/hip_runtime.h>

typedef __attribute__((ext_vector_type(2))) float v2f;
typedef __attribute__((ext_vector_type(4))) float v4f;
typedef __attribute__((ext_vector_type(8))) float v8f;

// Problem: inputs [Q=4096, D=128] f32, centers [C=256, D=128] f32
// out[c, q] = -||centers[c] - inputs[q]||^2
//           = 2*dot(centers[c], inputs[q]) - ||centers[c]||^2 - ||inputs[q]||^2
//
// GEMM via V_WMMA_F32_16X16X4_F32 (wave32). Per wave: 16 (C) x 64 (Q) tile.
// K axis processed in 8-column blocks with b128 loads: the WMMA K-reduction is
// permutation-invariant, and A/B lanes have identical (row, K-half) roles, so
// lanes 0-15 carry cols 8t..8t+3 and lanes 16-31 carry cols 8t+4..8t+7.
// (x,y) feeds one wmma step, (z,w) the next -> each column used exactly once.
// Row/col norms accumulate from the same fragments; shfl_xor(16) folds halves.

__global__ __launch_bounds__(256) void CentersDistance_wmma_kernel(
    const float* __restrict__ inputs,   // [Q, D]
    const float* __restrict__ centers,  // [C, D]
    float* __restrict__ out)            // [C, Q]
{
    constexpr int Q = 4096;
    constexpr int D = 128;

    const int lane = threadIdx.x & 31;
    const int wave = threadIdx.x >> 5;
    const int tile = blockIdx.x * 8 + wave;   // 1024 wave-tiles total

    const int c0 = (tile & 15) * 16;          // 16 tiles over C=256
    const int q0 = (tile >> 4) * 64;          // 64 tiles over Q=4096

    const int hi  = lane >> 4;                // K-half select
    const int l15 = lane & 15;

    // Each lane loads 4 consecutive floats: cols 8t + 4*hi .. +3
    const float* aptr = centers + (c0 + l15) * D + 4 * hi;
    const float* bptr = inputs + (q0 + l15) * D + 4 * hi;

    v8f acc0 = {}, acc1 = {}, acc2 = {}, acc3 = {};
    float asum = 0.f;
    float bsum0 = 0.f, bsum1 = 0.f, bsum2 = 0.f, bsum3 = 0.f;

    #pragma unroll 2
    for (int k = 0; k < D; k += 8) {
        v4f a  = *(const v4f*)(aptr + k);
        v4f b0 = *(const v4f*)(bptr + k);
        v4f b1 = *(const v4f*)(bptr + 16 * D + k);
        v4f b2 = *(const v4f*)(bptr + 32 * D + k);
        v4f b3 = *(const v4f*)(bptr + 48 * D + k);

        asum  += a.x  * a.x  + a.y  * a.y  + a.z  * a.z  + a.w  * a.w;
        bsum0 += b0.x * b0.x + b0.y * b0.y + b0.z * b0.z + b0.w * b0.w;
        bsum1 += b1.x * b1.x + b1.y * b1.y + b1.z * b1.z + b1.w * b1.w;
        bsum2 += b2.x * b2.x + b2.y * b2.y + b2.z * b2.z + b2.w * b2.w;
        bsum3 += b3.x * b3.x + b3.y * b3.y + b3.z * b3.z + b3.w * b3.w;

        v2f alo = {a.x, a.y},  ahi = {a.z, a.w};
        v2f l0  = {b0.x, b0.y}, h0 = {b0.z, b0.w};
        v2f l1  = {b1.x, b1.y}, h1 = {b1.z, b1.w};
        v2f l2  = {b2.x, b2.y}, h2 = {b2.z, b2.w};
        v2f l3  = {b3.x, b3.y}, h3 = {b3.z, b3.w};

        // 8 args: (neg_a, A, neg_b, B, c_mod, C, reuse_a, reuse_b)
        acc0 = __builtin_amdgcn_wmma_f32_16x16x4_f32(false, alo, false, l0,
                                                     (short)0, acc0, false, false);
        acc1 = __builtin_amdgcn_wmma_f32_16x16x4_f32(false, alo, false, l1,
                                                     (short)0, acc1, false, false);
        acc2 = __builtin_amdgcn_wmma_f32_16x16x4_f32(false, alo, false, l2,
                                                     (short)0, acc2, false, false);
        acc3 = __builtin_amdgcn_wmma_f32_16x16x4_f32(false, alo, false, l3,
                                                     (short)0, acc3, false, false);
        acc0 = __builtin_amdgcn_wmma_f32_16x16x4_f32(false, ahi, false, h0,
                                                     (short)0, acc0, false, false);
        acc1 = __builtin_amdgcn_wmma_f32_16x16x4_f32(false, ahi, false, h1,
                                                     (short)0, acc1, false, false);
        acc2 = __builtin_amdgcn_wmma_f32_16x16x4_f32(false, ahi, false, h2,
                                                     (short)0, acc2, false, false);
        acc3 = __builtin_amdgcn_wmma_f32_16x16x4_f32(false, ahi, false, h3,
                                                     (short)0, acc3, false, false);
    }

    // Fold the two half-wave K-groups: full norms per row/col.
    // After this, lane L holds cnorm[c0 + (L&15)] and qnorm[q0 + 16i + (L&15)].
    float cn  = asum  + __shfl_xor(asum,  16, 32);
    float qn0 = bsum0 + __shfl_xor(bsum0, 16, 32);
    float qn1 = bsum1 + __shfl_xor(bsum1, 16, 32);
    float qn2 = bsum2 + __shfl_xor(bsum2, 16, 32);
    float qn3 = bsum3 + __shfl_xor(bsum3, 16, 32);

    // C/D layout: VGPR j, lane L -> M = j + 8*(L>=16), N = L&15.
    // qn* is already aligned with N = L&15. Broadcast cnorm for this lane's rows.
    float crow[8];
    #pragma unroll
    for (int j = 0; j < 8; ++j)
        crow[j] = __shfl(cn, j + 8 * hi, 32);

    #pragma unroll
    for (int j = 0; j < 8; ++j) {
        const int row = c0 + j + 8 * hi;
        float* p = out + row * Q + q0 + l15;
        p[0]  = 2.f * acc0[j] - crow[j] - qn0;
        p[16] = 2.f * acc1[j] - crow[j] - qn1;
        p[32] = 2.f * acc2[j] - crow[j] - qn2;
        p[48] = 2.f * acc3[j] - crow[j] - qn3;
    }
}

extern "C" void kernel_launch(void* const* d_in, const int* in_sizes, int n_in,
                              void* d_out, int out_size, void* d_ws, size_t ws_size,
                              hipStream_t stream) {
    (void)in_sizes; (void)n_in; (void)out_size; (void)d_ws; (void)ws_size;
    const float* inputs  = (const float*)d_in[0];   // [4096, 128]
    const float* centers = (const float*)d_in[1];   // [256, 128]
    float* out = (float*)d_out;                     // [256, 4096]

    // 1024 wave-tiles (16 over C, 64 over Q), 8 waves per 256-thread block.
    CentersDistance_wmma_kernel<<<128, 256, 0, stream>>>(inputs, centers, out);
}